// CrossMambaBlock_88699664597216
// MI455X (gfx1250) — compile-verified
//
#include <hip/hip_runtime.h>
#include <hip/hip_bf16.h>
#include <stdint.h>

// ---------------- problem constants ----------------
#define DMOD   256          // D_MODEL
#define DI     512          // D_INNER
#define DS_N   16           // D_STATE
#define DTR    16           // DT_RANK
#define DCONV  4
#define NSEQ   128          // B*H == B*W sequences
#define LSEQ   64
#define MROWS  (NSEQ*LSEQ)  // 8192
#define NXZ    (2*DI)       // 1024
#define NDBL   (DTR+2*DS_N) // 48

typedef __attribute__((ext_vector_type(2))) float v2f;
typedef __attribute__((ext_vector_type(8))) float v8f;

// D = A(16x4,f32) * B(4x16,f32) + C ; fp32 throughout (precision-preserving).
__device__ __forceinline__ v8f wmma4(v2f a, v2f b, v8f c) {
#if defined(__HIP_DEVICE_COMPILE__)
  return __builtin_amdgcn_wmma_f32_16x16x4_f32(false, a, false, b, (short)0, c,
                                               false, false);
#else
  (void)a; (void)b;
  return c;
#endif
}

// Async Global->LDS 16B copy (CDNA5 TDM-adjacent path, ASYNCcnt-tracked).
// LDS address comes from the vdst VGPR; low 32 bits of the flat shared pointer
// are the wave-relative LDS byte offset.
__device__ __forceinline__ void async_ld_b128(void* lds_ptr, const void* gptr) {
#if defined(__HIP_DEVICE_COMPILE__)
  unsigned lds_off = (unsigned)(uintptr_t)lds_ptr;
  asm volatile("global_load_async_to_lds_b128 %0, %1, off"
               :: "v"(lds_off), "v"(gptr) : "memory");
#else
  (void)lds_ptr; (void)gptr;
#endif
}

__device__ __forceinline__ void async_wait0() {
#if defined(__HIP_DEVICE_COMPILE__)
  asm volatile("s_wait_asynccnt 0x0" ::: "memory");
#endif
}

// ------------------------------------------------------------------
// K1: xz = u @ in_w.T   (8192x256)x(256x1024), both mambas (blockIdx.z)
// 8 waves/block, each wave one 16x16 tile; A tile async-staged in LDS.
// ------------------------------------------------------------------
__global__ __launch_bounds__(256) void k_inproj(const float* __restrict__ x,
                                                const float* __restrict__ w1,
                                                const float* __restrict__ w2,
                                                float* __restrict__ xz1,
                                                float* __restrict__ xz2) {
  __shared__ __align__(16) float As[16 * 260];
  const int m0 = blockIdx.x * 16;
  const int n0 = blockIdx.y * 128 + (threadIdx.x >> 5) * 16;
  const float* __restrict__ W = blockIdx.z ? w2 : w1;
  float* __restrict__ out = blockIdx.z ? xz2 : xz1;

  // async-coalesced load of the 16x256 A tile: 1024 x b128 chunks
  #pragma unroll
  for (int i = 0; i < 4; ++i) {
    int idx = threadIdx.x + i * 256;
    int r = idx >> 6, c = (idx & 63) << 2;
    async_ld_b128(&As[r * 260 + c], &x[(m0 + r) * DMOD + c]);
  }
  async_wait0();
  __syncthreads();

  const int lane = threadIdx.x & 31;
  const int mr = lane & 15;
  const int kh = (lane >> 4) << 1;
  const float* __restrict__ Wp = W + (size_t)(n0 + mr) * DMOD;
  __builtin_prefetch(Wp, 0, 1);
  v8f acc = {};
  for (int k0 = 0; k0 < DMOD; k0 += 4) {
    v2f a = *(const v2f*)&As[mr * 260 + k0 + kh];
    v2f b = *(const v2f*)&Wp[k0 + kh];
    acc = wmma4(a, b, acc);
  }
  float* __restrict__ Cp = out + (size_t)m0 * NXZ + n0;
  const int col = mr, rh = (lane >> 4) * 8;
  #pragma unroll
  for (int v = 0; v < 8; ++v) Cp[(size_t)(v + rh) * NXZ + col] = acc[v];
}

// ------------------------------------------------------------------
// K2: depthwise causal conv (K=4) over L + SiLU, reading x-half of xz
// ------------------------------------------------------------------
__global__ __launch_bounds__(256) void k_conv_silu(
    const float* __restrict__ xz1, const float* __restrict__ xz2,
    const float* __restrict__ cw1, const float* __restrict__ cb1,
    const float* __restrict__ cw2, const float* __restrict__ cb2,
    float* __restrict__ xc1, float* __restrict__ xc2) {
  const float* __restrict__ xz = blockIdx.y ? xz2 : xz1;
  const float* __restrict__ cw = blockIdx.y ? cw2 : cw1;
  const float* __restrict__ cb = blockIdx.y ? cb2 : cb1;
  float* __restrict__ xc = blockIdx.y ? xc2 : xc1;

  int idx = blockIdx.x * 256 + threadIdx.x;  // over MROWS*DI
  int d = idx & (DI - 1);
  int row = idx >> 9;              // s*64 + l
  int l = row & (LSEQ - 1);
  float acc = cb[d];
  #pragma unroll
  for (int k = 0; k < DCONV; ++k) {
    int li = l + k - (DCONV - 1);
    float xv = (li >= 0) ? xz[(size_t)(row + k - (DCONV - 1)) * NXZ + d] : 0.f;
    acc = fmaf(xv, cw[d * DCONV + k], acc);
  }
  xc[idx] = acc / (1.f + expf(-acc));  // silu
}

// ------------------------------------------------------------------
// K3: xdbl = xc @ xp_w.T   (8192x512)x(512x48); one wave per 16x16 tile
// ------------------------------------------------------------------
__global__ __launch_bounds__(32) void k_xproj(const float* __restrict__ xc1,
                                              const float* __restrict__ xc2,
                                              const float* __restrict__ w1,
                                              const float* __restrict__ w2,
                                              float* __restrict__ xd1,
                                              float* __restrict__ xd2) {
  const float* __restrict__ A = blockIdx.z ? xc2 : xc1;
  const float* __restrict__ W = blockIdx.z ? w2 : w1;
  float* __restrict__ C = blockIdx.z ? xd2 : xd1;
  const int m0 = blockIdx.x * 16;
  const int n0 = blockIdx.y * 16;
  const int lane = threadIdx.x;
  const int mr = lane & 15, kh = (lane >> 4) << 1;
  const float* __restrict__ Ap = A + (size_t)(m0 + mr) * DI;
  const float* __restrict__ Wp = W + (size_t)(n0 + mr) * DI;
  v8f acc = {};
  for (int k0 = 0; k0 < DI; k0 += 4) {
    v2f a = *(const v2f*)&Ap[k0 + kh];
    v2f b = *(const v2f*)&Wp[k0 + kh];
    acc = wmma4(a, b, acc);
  }
  float* __restrict__ Cp = C + (size_t)m0 * NDBL + n0;
  const int col = mr, rh = (lane >> 4) * 8;
  #pragma unroll
  for (int v = 0; v < 8; ++v) Cp[(size_t)(v + rh) * NDBL + col] = acc[v];
}

// ------------------------------------------------------------------
// K4: dt = softplus(xdbl[:, :16] @ dt_w.T + dt_b)   K=16 WMMA + epilogue
// ------------------------------------------------------------------
__global__ __launch_bounds__(256) void k_dtproj(const float* __restrict__ xd1,
                                                const float* __restrict__ xd2,
                                                const float* __restrict__ w1,
                                                const float* __restrict__ b1,
                                                const float* __restrict__ w2,
                                                const float* __restrict__ b2,
                                                float* __restrict__ dt1,
                                                float* __restrict__ dt2) {
  const float* __restrict__ A = blockIdx.z ? xd2 : xd1;
  const float* __restrict__ W = blockIdx.z ? w2 : w1;
  const float* __restrict__ bias = blockIdx.z ? b2 : b1;
  float* __restrict__ C = blockIdx.z ? dt2 : dt1;
  const int m0 = blockIdx.x * 16;
  const int n0 = blockIdx.y * 128 + (threadIdx.x >> 5) * 16;
  const int lane = threadIdx.x & 31;
  const int mr = lane & 15, kh = (lane >> 4) << 1;
  const float* __restrict__ Ap = A + (size_t)(m0 + mr) * NDBL;  // cols 0..15
  const float* __restrict__ Wp = W + (size_t)(n0 + mr) * DTR;
  v8f acc = {};
  #pragma unroll
  for (int k0 = 0; k0 < DTR; k0 += 4) {
    v2f a = *(const v2f*)&Ap[k0 + kh];
    v2f b = *(const v2f*)&Wp[k0 + kh];
    acc = wmma4(a, b, acc);
  }
  const int col = mr, rh = (lane >> 4) * 8;
  const float bv = bias[n0 + col];
  float* __restrict__ Cp = C + (size_t)m0 * DI + n0;
  #pragma unroll
  for (int v = 0; v < 8; ++v) {
    float t = acc[v] + bv;
    float sp = (t > 20.f) ? t : log1pf(expf(t));  // softplus
    Cp[(size_t)(v + rh) * DI + col] = sp;
  }
}

// ------------------------------------------------------------------
// K5: selective scan. One thread per (sequence, channel); 16-wide state
// in registers; fused  y = (scan + x*D) * silu(z)  (writes over dt buf).
// ------------------------------------------------------------------
__global__ __launch_bounds__(256) void k_scan(
    const float* __restrict__ xz1, const float* __restrict__ xz2,
    const float* __restrict__ xc1, const float* __restrict__ xc2,
    const float* __restrict__ xd1, const float* __restrict__ xd2,
    const float* __restrict__ dt1, const float* __restrict__ dt2,
    const float* __restrict__ Al1, const float* __restrict__ Dp1,
    const float* __restrict__ Al2, const float* __restrict__ Dp2,
    float* __restrict__ yg1, float* __restrict__ yg2) {
  const float* __restrict__ xz = blockIdx.y ? xz2 : xz1;
  const float* __restrict__ xc = blockIdx.y ? xc2 : xc1;
  const float* __restrict__ xd = blockIdx.y ? xd2 : xd1;
  const float* __restrict__ dt = blockIdx.y ? dt2 : dt1;
  const float* __restrict__ Al = blockIdx.y ? Al2 : Al1;
  const float* __restrict__ Dp = blockIdx.y ? Dp2 : Dp1;
  float* __restrict__ yg = blockIdx.y ? yg2 : yg1;

  const int idx = blockIdx.x * 256 + threadIdx.x;  // over NSEQ*DI
  const int d = idx & (DI - 1);
  const int s = idx >> 9;
  float Ac[DS_N], h[DS_N];
  #pragma unroll
  for (int n = 0; n < DS_N; ++n) {
    Ac[n] = -expf(Al[d * DS_N + n]);
    h[n] = 0.f;
  }
  const float Dv = Dp[d];
  for (int l = 0; l < LSEQ; ++l) {
    const size_t row = (size_t)s * LSEQ + l;
    const float dtv = dt[row * DI + d];
    const float xv = xc[row * DI + d];
    const float zv = xz[row * NXZ + DI + d];
    const float* __restrict__ bc = xd + row * NDBL;  // B at +16, C at +32
    const float dx = dtv * xv;
    float y = 0.f;
    #pragma unroll
    for (int n = 0; n < DS_N; ++n) {
      float da = expf(dtv * Ac[n]);
      h[n] = fmaf(da, h[n], dx * bc[DTR + n]);
      y = fmaf(h[n], bc[DTR + DS_N + n], y);
    }
    y = fmaf(xv, Dv, y);
    const float sig = 1.f / (1.f + expf(-zv));
    yg[row * DI + d] = y * (zv * sig);   // safe in-place over dt[row,d]
  }
}

// ------------------------------------------------------------------
// K6: y @ out_w.T  (8192x512)x(512x256) and scatter straight into the
// cat layout (bp, channel):  element(r,f) -> cat[(b*4096+(r&15)*256+f),
// mOff + ((r>>4)&255)]   (derived from the reshape semantics)
// ------------------------------------------------------------------
__global__ __launch_bounds__(256) void k_outproj(const float* __restrict__ yg1,
                                                 const float* __restrict__ yg2,
                                                 const float* __restrict__ w1,
                                                 const float* __restrict__ w2,
                                                 float* __restrict__ cat) {
  __shared__ __align__(16) float As[16 * 516];
  const float* __restrict__ A = blockIdx.z ? yg2 : yg1;
  const float* __restrict__ W = blockIdx.z ? w2 : w1;
  const int m0 = blockIdx.x * 16;
  const int n0 = blockIdx.y * 128 + (threadIdx.x >> 5) * 16;
  // async stage of 16x512 A tile: 2048 x b128 chunks
  #pragma unroll
  for (int i = 0; i < 8; ++i) {
    int idx = threadIdx.x + i * 256;
    int r = idx >> 7, c = (idx & 127) << 2;
    async_ld_b128(&As[r * 516 + c], &A[(size_t)(m0 + r) * DI + c]);
  }
  async_wait0();
  __syncthreads();

  const int lane = threadIdx.x & 31;
  const int mr = lane & 15, kh = (lane >> 4) << 1;
  const float* __restrict__ Wp = W + (size_t)(n0 + mr) * DI;
  __builtin_prefetch(Wp, 0, 1);
  v8f acc = {};
  for (int k0 = 0; k0 < DI; k0 += 4) {
    v2f a = *(const v2f*)&As[mr * 516 + k0 + kh];
    v2f b = *(const v2f*)&Wp[k0 + kh];
    acc = wmma4(a, b, acc);
  }
  const int b = m0 >> 12;
  const int ch = (m0 >> 4) & 255;
  const int mOff = blockIdx.z ? 256 : 0;
  const int col = n0 + mr;
  const int rh = (lane >> 4) * 8;
  #pragma unroll
  for (int v = 0; v < 8; ++v) {
    int i = v + rh;                            // r & 15 within tile
    size_t bp = (size_t)b * 4096 + (size_t)i * 256 + col;
    cat[bp * 512 + mOff + ch] = acc[v];
  }
}

// ------------------------------------------------------------------
// K7: out = cat @ cat_w.T + cat_b, scattered to (b,o,h,w) + residual x
// ------------------------------------------------------------------
__global__ __launch_bounds__(256) void k_catconv(const float* __restrict__ cat,
                                                 const float* __restrict__ W,
                                                 const float* __restrict__ bias,
                                                 const float* __restrict__ xin,
                                                 float* __restrict__ out) {
  __shared__ __align__(16) float As[16 * 516];
  const int m0 = blockIdx.x * 16;
  const int n0 = blockIdx.y * 128 + (threadIdx.x >> 5) * 16;
  #pragma unroll
  for (int i = 0; i < 8; ++i) {
    int idx = threadIdx.x + i * 256;
    int r = idx >> 7, c = (idx & 127) << 2;
    async_ld_b128(&As[r * 516 + c], &cat[(size_t)(m0 + r) * 512 + c]);
  }
  async_wait0();
  __syncthreads();

  const int lane = threadIdx.x & 31;
  const int mr = lane & 15, kh = (lane >> 4) << 1;
  const float* __restrict__ Wp = W + (size_t)(n0 + mr) * 512;
  v8f acc = {};
  for (int k0 = 0; k0 < 512; k0 += 4) {
    v2f a = *(const v2f*)&As[mr * 516 + k0 + kh];
    v2f b = *(const v2f*)&Wp[k0 + kh];
    acc = wmma4(a, b, acc);
  }
  const int col = n0 + mr;  // output channel o
  const int rh = (lane >> 4) * 8;
  const float bv = bias[col];
  #pragma unroll
  for (int v = 0; v < 8; ++v) {
    int r = m0 + v + rh;  // bp = b*4096 + p
    int b = r >> 12;
    int p = r & 4095;
    size_t oidx = (size_t)b * 1048576 + (size_t)col * 4096 + p;
    out[oidx] = acc[v] + bv + xin[oidx];
  }
}

// ------------------------------------------------------------------
extern "C" void kernel_launch(void* const* d_in, const int* in_sizes, int n_in,
                              void* d_out, int out_size, void* d_ws,
                              size_t ws_size, hipStream_t stream) {
  const float* x      = (const float*)d_in[0];
  const float* in_w1  = (const float*)d_in[1];
  const float* cw1    = (const float*)d_in[2];
  const float* cb1    = (const float*)d_in[3];
  const float* xp_w1  = (const float*)d_in[4];
  const float* dt_w1  = (const float*)d_in[5];
  const float* dt_b1  = (const float*)d_in[6];
  const float* Alog1  = (const float*)d_in[7];
  const float* D1     = (const float*)d_in[8];
  const float* out_w1 = (const float*)d_in[9];
  const float* in_w2  = (const float*)d_in[10];
  const float* cw2    = (const float*)d_in[11];
  const float* cb2    = (const float*)d_in[12];
  const float* xp_w2  = (const float*)d_in[13];
  const float* dt_w2  = (const float*)d_in[14];
  const float* dt_b2  = (const float*)d_in[15];
  const float* Alog2  = (const float*)d_in[16];
  const float* D2     = (const float*)d_in[17];
  const float* out_w2 = (const float*)d_in[18];
  const float* cat_w  = (const float*)d_in[19];
  const float* cat_b  = (const float*)d_in[20];

  float* ws  = (float*)d_ws;
  float* xz1 = ws;                              // 8192*1024
  float* xz2 = xz1 + (size_t)MROWS * NXZ;       // 8192*1024
  float* xc1 = xz2 + (size_t)MROWS * NXZ;       // 8192*512
  float* xc2 = xc1 + (size_t)MROWS * DI;        // 8192*512
  float* xd1 = xc2 + (size_t)MROWS * DI;        // 8192*48
  float* xd2 = xd1 + (size_t)MROWS * NDBL;      // 8192*48
  float* dt1 = xd2 + (size_t)MROWS * NDBL;      // 8192*512
  float* dt2 = dt1 + (size_t)MROWS * DI;        // 8192*512
  float* cat = dt2 + (size_t)MROWS * DI;        // 8192*512
  float* yg1 = dt1;  // scan overwrites dt in place (per-thread, race-free)
  float* yg2 = dt2;
  float* out = (float*)d_out;

  k_inproj<<<dim3(MROWS / 16, NXZ / 128, 2), dim3(256), 0, stream>>>(
      x, in_w1, in_w2, xz1, xz2);
  k_conv_silu<<<dim3((MROWS * DI) / 256, 2), dim3(256), 0, stream>>>(
      xz1, xz2, cw1, cb1, cw2, cb2, xc1, xc2);
  k_xproj<<<dim3(MROWS / 16, NDBL / 16, 2), dim3(32), 0, stream>>>(
      xc1, xc2, xp_w1, xp_w2, xd1, xd2);
  k_dtproj<<<dim3(MROWS / 16, DI / 128, 2), dim3(256), 0, stream>>>(
      xd1, xd2, dt_w1, dt_b1, dt_w2, dt_b2, dt1, dt2);
  k_scan<<<dim3((NSEQ * DI) / 256, 2), dim3(256), 0, stream>>>(
      xz1, xz2, xc1, xc2, xd1, xd2, dt1, dt2, Alog1, D1, Alog2, D2, yg1, yg2);
  k_outproj<<<dim3(MROWS / 16, 256 / 128, 2), dim3(256), 0, stream>>>(
      yg1, yg2, out_w1, out_w2, cat);
  k_catconv<<<dim3(MROWS / 16, 256 / 128), dim3(256), 0, stream>>>(
      cat, cat_w, cat_b, x, out);
}